// LossRegressionGaussianNoCorrelations_11312943857917
// MI455X (gfx1250) — compile-verified
//
#include <hip/hip_runtime.h>
#include <math.h>

// ---------------------------------------------------------------------------
// Negative-ELBO reduction for a Bayesian regression loss on MI455X (gfx1250).
// Pure streaming problem: 320 MB read @ 23.3 TB/s => ~13.7 us floor.
// Strategy: b128 non-temporal loads, grid-stride fp32 accumulation,
// WMMA-based (v_wmma_f32_16x16x4_f32) intra-wave reduction, deterministic
// two-pass block reduction with fp64 final accumulation.
// ---------------------------------------------------------------------------

#define HALF_LOG_2PI 0.9189385332046727

typedef float v2f __attribute__((ext_vector_type(2)));
typedef float v4f __attribute__((ext_vector_type(4)));
typedef float v8f __attribute__((ext_vector_type(8)));

#define NBLOCKS  4096
#define NTHREADS 256
#define WAVES_PER_BLOCK (NTHREADS / 32)

// Full 32-lane wave sum using one f32 WMMA.
// A-matrix (16x4 f32): lane l in [0,15] supplies A[l][0..1] = acc_l (2 VGPRs),
// lane l in [16,31] supplies A[l-16][2..3] = acc_l.  B = 0.5 everywhere, so
// D[m][n] = 0.5*(2*acc_m + 2*acc_{m+16}) = acc_m + acc_{m+16}.
// C/D layout: lanes 0-15 hold rows 0..7 in d[0..7], lanes 16-31 hold rows 8..15.
// Summing the 8 components gives each half-wave's row-sum; one shfl_xor(16)
// completes the full wave reduction.  EXEC must be all-ones (callers are
// convergent here).
__device__ __forceinline__ float wave_reduce_wmma(float acc) {
    v2f a;  a[0] = acc;   a[1] = acc;
    v2f b;  b[0] = 0.5f;  b[1] = 0.5f;
    v8f c = {};
    v8f d = __builtin_amdgcn_wmma_f32_16x16x4_f32(
        /*neg_a=*/false, a, /*neg_b=*/false, b,
        /*c_mod=*/(short)0, c, /*reuse_a=*/false, /*reuse_b=*/false);
    float h = ((d[0] + d[1]) + (d[2] + d[3])) + ((d[4] + d[5]) + (d[6] + d[7]));
    h += __shfl_xor(h, 16, 32);   // combine the two half-wave row-sums
    return h;
}

__global__ void __launch_bounds__(NTHREADS)
elbo_partial_kernel(const float* __restrict__ noisy,
                    const float* __restrict__ mu_w,
                    const float* __restrict__ sig_w,
                    const float* __restrict__ mu_p,
                    const float* __restrict__ sig_p,   // scalar (1 element)
                    const float* __restrict__ y,
                    float* __restrict__ partial,
                    int n_w, int n_b) {
    const float inv_sp = 1.0f / sig_p[0];
    const int   tid    = blockIdx.x * blockDim.x + threadIdx.x;
    const int   stride = gridDim.x * blockDim.x;

    float acc = 0.0f;

    // ---- weight streams: variational - prior (per-element, constants cancel)
    {
        const int n4 = n_w >> 2;
        const v4f* x4 = (const v4f*)noisy;
        const v4f* m4 = (const v4f*)mu_w;
        const v4f* s4 = (const v4f*)sig_w;
        for (int i = tid; i < n4; i += stride) {
            v4f xv = __builtin_nontemporal_load(x4 + i);
            v4f mv = __builtin_nontemporal_load(m4 + i);
            v4f sv = __builtin_nontemporal_load(s4 + i);
#pragma unroll
            for (int j = 0; j < 4; ++j) {
                float x = xv[j], m = mv[j], s = sv[j];
                float zq = (x - m) / s;
                acc += 0.5f * x * x - 0.5f * zq * zq - __logf(s);
            }
        }
        for (int i = (n4 << 2) + tid; i < n_w; i += stride) {  // tail
            float x = noisy[i], m = mu_w[i], s = sig_w[i];
            float zq = (x - m) / s;
            acc += 0.5f * x * x - 0.5f * zq * zq - __logf(s);
        }
    }

    // ---- batch streams: minus likelihood (per-element quadratic part)
    {
        const int n4 = n_b >> 2;
        const v4f* p4  = (const v4f*)mu_p;
        const v4f* yy4 = (const v4f*)y;
        for (int i = tid; i < n4; i += stride) {
            v4f pv = __builtin_nontemporal_load(p4 + i);
            v4f yv = __builtin_nontemporal_load(yy4 + i);
#pragma unroll
            for (int j = 0; j < 4; ++j) {
                float zl = (yv[j] - pv[j]) * inv_sp;
                acc += 0.5f * zl * zl;
            }
        }
        for (int i = (n4 << 2) + tid; i < n_b; i += stride) {  // tail
            float zl = (y[i] - mu_p[i]) * inv_sp;
            acc += 0.5f * zl * zl;
        }
    }

    // ---- wave reduction via WMMA (all threads convergent here, EXEC all-1s)
    float wsum = wave_reduce_wmma(acc);

    __shared__ float lds[WAVES_PER_BLOCK];
    const int lane = threadIdx.x & 31;
    const int wid  = threadIdx.x >> 5;
    if (lane == 0) lds[wid] = wsum;
    __syncthreads();
    if (threadIdx.x == 0) {
        float s = 0.0f;
#pragma unroll
        for (int w = 0; w < WAVES_PER_BLOCK; ++w) s += lds[w];  // fixed order
        partial[blockIdx.x] = s;
    }
}

__global__ void __launch_bounds__(NTHREADS)
elbo_final_kernel(const float* __restrict__ partial,
                  const float* __restrict__ sig_p,
                  float* __restrict__ out,
                  int nblocks, long long n_batch) {
    __shared__ double lds[NTHREADS];
    double s = 0.0;
    for (int i = threadIdx.x; i < nblocks; i += blockDim.x)
        s += (double)partial[i];                       // fixed assignment
    lds[threadIdx.x] = s;
    __syncthreads();
    for (int off = NTHREADS / 2; off > 0; off >>= 1) { // deterministic tree
        if (threadIdx.x < off) lds[threadIdx.x] += lds[threadIdx.x + off];
        __syncthreads();
    }
    if (threadIdx.x == 0) {
        // constant term from the likelihood: + N_batch * (log(sigma_p) + 0.5*log(2*pi))
        double c = (double)n_batch * (log((double)sig_p[0]) + HALF_LOG_2PI);
        out[0] = (float)(lds[0] + c);
    }
}

extern "C" void kernel_launch(void* const* d_in, const int* in_sizes, int n_in,
                              void* d_out, int out_size, void* d_ws, size_t ws_size,
                              hipStream_t stream) {
    const float* noisy = (const float*)d_in[0];
    const float* mu_w  = (const float*)d_in[1];
    const float* sig_w = (const float*)d_in[2];
    const float* mu_p  = (const float*)d_in[3];
    const float* sig_p = (const float*)d_in[4];   // single-element scalar
    const float* y     = (const float*)d_in[5];

    const int n_w = in_sizes[0];
    const int n_b = in_sizes[5];

    float* partial = (float*)d_ws;                // NBLOCKS floats = 16 KB

    elbo_partial_kernel<<<NBLOCKS, NTHREADS, 0, stream>>>(
        noisy, mu_w, sig_w, mu_p, sig_p, y, partial, n_w, n_b);

    elbo_final_kernel<<<1, NTHREADS, 0, stream>>>(
        partial, sig_p, (float*)d_out, NBLOCKS, (long long)n_b);
}